// multihead_gat_3788161155708
// MI455X (gfx1250) — compile-verified
//
#include <hip/hip_runtime.h>
#include <hip/hip_bf16.h>

typedef __attribute__((ext_vector_type(2))) float v2f;
typedef __attribute__((ext_vector_type(8))) float v8f;

#define N_NODES   4096
#define IN_DIM    512
#define OUT_DIM   64
#define NEG_SLOPE 0.01f
#define MASK_VAL  (-1000000000000.0f)

#define JSPLIT    4
#define JCHUNK    (N_NODES / JSPLIT)   // 1024 columns per j-split

// ---------------------------------------------------------------------------
// Kernel 1: h = X @ W   (4096x512 @ 512x64), fp32 WMMA 16x16x4.
// One wave per 16x16 output tile; 4 waves/block cover the 64 output columns.
// A-fragment (16x4 f32): lanes 0-15 hold row M=lane, K=k..k+1; lanes 16-31
// hold row M=lane-16, K=k+2..k+3 (ISA 7.12.2 32-bit A layout).
// ---------------------------------------------------------------------------
__global__ __launch_bounds__(128) void gat_h_gemm(const float* __restrict__ X,
                                                  const float* __restrict__ W,
                                                  float* __restrict__ h) {
  const int lane = threadIdx.x & 31;
  const int wave = threadIdx.x >> 5;
  const int hl   = lane >> 4;     // 0 or 1
  const int r    = lane & 15;
  const int m0   = blockIdx.x * 16;
  const int n0   = wave * 16;

  const float* __restrict__ Xrow = X + (m0 + r) * IN_DIM;

  v8f c = {};
  for (int k = 0; k < IN_DIM; k += 4) {
    const int kk = k + 2 * hl;                    // even -> 8B aligned
    const v2f a = *(const v2f*)(Xrow + kk);       // global_load_b64
    v2f b;
    b.x = W[kk * OUT_DIM + n0 + r];
    b.y = W[(kk + 1) * OUT_DIM + n0 + r];
    c = __builtin_amdgcn_wmma_f32_16x16x4_f32(false, a, false, b, (short)0, c,
                                              false, false);
  }
#pragma unroll
  for (int v = 0; v < 8; ++v) {
    h[(m0 + v + 8 * hl) * OUT_DIM + n0 + r] = c[v];
  }
}

// ---------------------------------------------------------------------------
// Kernel 2: s1[i] = h[i] . a[0:64],  s2[i] = h[i] . a[64:128]   (tiny)
// ---------------------------------------------------------------------------
__global__ __launch_bounds__(256) void gat_scores(const float* __restrict__ h,
                                                  const float* __restrict__ a,
                                                  float* __restrict__ s1,
                                                  float* __restrict__ s2) {
  const int i = blockIdx.x * blockDim.x + threadIdx.x;
  if (i >= N_NODES) return;
  const float* __restrict__ hr = h + i * OUT_DIM;
  float acc1 = 0.f, acc2 = 0.f;
#pragma unroll 8
  for (int d = 0; d < OUT_DIM; ++d) {
    const float hv = hr[d];
    acc1 += hv * a[d];
    acc2 += hv * a[OUT_DIM + d];
  }
  s1[i] = acc1;
  s2[i] = acc2;
}

// ---------------------------------------------------------------------------
// Kernel 3: per-row softmax stats (max, sum of exp) — one wave32 per row.
// Each lane owns 4 consecutive columns -> int4/float4 (b128) streaming loads.
// Online (max,sum) merge, then wave32 butterfly combine. Reads adj once.
// ---------------------------------------------------------------------------
__global__ __launch_bounds__(256) void gat_softmax_stats(
    const int* __restrict__ adj, const float* __restrict__ s1,
    const float* __restrict__ s2, float* __restrict__ rowmax,
    float* __restrict__ rowsum) {
  const int lane = threadIdx.x & 31;
  const int wave = threadIdx.x >> 5;
  const int row  = blockIdx.x * 8 + wave;

  const float s1r = s1[row];
  const int* __restrict__ adjRow = adj + row * N_NODES;

  float m = -__builtin_inff();
  float l = 0.f;
  for (int j4 = lane * 4; j4 < N_NODES; j4 += 128) {
    const int4   av  = *(const int4*)(adjRow + j4);   // 16B aligned
    const float4 s2v = *(const float4*)(s2 + j4);     // 16B aligned
    const int   ai[4] = {av.x, av.y, av.z, av.w};
    const float sv[4] = {s2v.x, s2v.y, s2v.z, s2v.w};
#pragma unroll
    for (int t = 0; t < 4; ++t) {
      float e = s1r + sv[t];
      e = (e >= 0.f) ? e : NEG_SLOPE * e;
      const float sc = (ai[t] > 0) ? e : MASK_VAL;
      const float mn = fmaxf(m, sc);
      l = l * __expf(m - mn) + __expf(sc - mn);
      m = mn;
    }
  }
#pragma unroll
  for (int off = 16; off >= 1; off >>= 1) {
    const float mo = __shfl_xor(m, off, 32);
    const float lo = __shfl_xor(l, off, 32);
    const float mn = fmaxf(m, mo);
    l = l * __expf(m - mn) + lo * __expf(mo - mn);
    m = mn;
  }
  if (lane == 0) {
    rowmax[row] = m;
    rowsum[row] = l;
  }
}

// ---------------------------------------------------------------------------
// Kernel 4: partial[js] = softmax(scores)[:, js-range] @ h[js-range, :].
// Wave owns 16 rows x 64 cols (4 fp32 WMMA accumulator tiles); the j (K)
// dimension is split JSPLIT ways across blockIdx.y for 4x more waves in
// flight on the adj/L2 stream. Probabilities (already scaled by 1/l_i) are
// generated on the fly directly in the WMMA A-fragment layout.
// ---------------------------------------------------------------------------
__global__ __launch_bounds__(128) void gat_attn_gemm(
    const int* __restrict__ adj, const float* __restrict__ s1,
    const float* __restrict__ s2, const float* __restrict__ rowmax,
    const float* __restrict__ rowsum, const float* __restrict__ h,
    float* __restrict__ partial) {
  const int lane = threadIdx.x & 31;
  const int wave = threadIdx.x >> 5;
  const int hl   = lane >> 4;
  const int r    = lane & 15;
  const int m0   = (blockIdx.x * 4 + wave) * 16;
  const int jb   = blockIdx.y * JCHUNK;

  const float s1r  = s1[m0 + r];
  const float mr   = rowmax[m0 + r];
  const float invl = 1.0f / rowsum[m0 + r];
  const int* __restrict__ adjRow = adj + (m0 + r) * N_NODES;

  v8f c0 = {}, c1 = {}, c2 = {}, c3 = {};

  for (int jc = jb; jc < jb + JCHUNK; jc += 4) {
    const int j0 = jc + 2 * hl;                   // even -> 8B aligned
    const int2 av  = *(const int2*)(adjRow + j0); // global_load_b64
    const v2f  s2v = *(const v2f*)(s2 + j0);      // global_load_b64
    float e0 = s1r + s2v.x;
    float e1 = s1r + s2v.y;
    e0 = (e0 >= 0.f) ? e0 : NEG_SLOPE * e0;
    e1 = (e1 >= 0.f) ? e1 : NEG_SLOPE * e1;
    const float sc0 = (av.x > 0) ? e0 : MASK_VAL;
    const float sc1 = (av.y > 0) ? e1 : MASK_VAL;
    v2f A;
    A.x = __expf(sc0 - mr) * invl;
    A.y = __expf(sc1 - mr) * invl;

    const float* __restrict__ h0 = h + j0 * OUT_DIM;
    const float* __restrict__ h1 = h0 + OUT_DIM;
    v2f b;
    b.x = h0[r];
    b.y = h1[r];
    c0 = __builtin_amdgcn_wmma_f32_16x16x4_f32(false, A, false, b, (short)0,
                                               c0, false, false);
    b.x = h0[16 + r];
    b.y = h1[16 + r];
    c1 = __builtin_amdgcn_wmma_f32_16x16x4_f32(false, A, false, b, (short)0,
                                               c1, false, false);
    b.x = h0[32 + r];
    b.y = h1[32 + r];
    c2 = __builtin_amdgcn_wmma_f32_16x16x4_f32(false, A, false, b, (short)0,
                                               c2, false, false);
    b.x = h0[48 + r];
    b.y = h1[48 + r];
    c3 = __builtin_amdgcn_wmma_f32_16x16x4_f32(false, A, false, b, (short)0,
                                               c3, false, false);
  }

  float* __restrict__ p = partial + (size_t)blockIdx.y * N_NODES * OUT_DIM;
#pragma unroll
  for (int v = 0; v < 8; ++v) {
    const int row = m0 + v + 8 * hl;
    p[row * OUT_DIM + r]      = c0[v];
    p[row * OUT_DIM + 16 + r] = c1[v];
    p[row * OUT_DIM + 32 + r] = c2[v];
    p[row * OUT_DIM + 48 + r] = c3[v];
  }
}

// ---------------------------------------------------------------------------
// Kernel 5: out = sum over JSPLIT partials (deterministic reduction).
// ---------------------------------------------------------------------------
__global__ __launch_bounds__(256) void gat_reduce(
    const float* __restrict__ partial, float* __restrict__ out) {
  const int idx = blockIdx.x * blockDim.x + threadIdx.x;
  if (idx >= N_NODES * OUT_DIM) return;
  float acc = 0.f;
#pragma unroll
  for (int s = 0; s < JSPLIT; ++s) {
    acc += partial[(size_t)s * N_NODES * OUT_DIM + idx];
  }
  out[idx] = acc;
}

// ---------------------------------------------------------------------------
extern "C" void kernel_launch(void* const* d_in, const int* in_sizes, int n_in,
                              void* d_out, int out_size, void* d_ws,
                              size_t ws_size, hipStream_t stream) {
  const float* X   = (const float*)d_in[0];  // 4096 x 512
  const int*   adj = (const int*)d_in[1];    // 4096 x 4096
  const float* W   = (const float*)d_in[2];  // 512 x 64
  const float* a   = (const float*)d_in[3];  // 128
  float* out = (float*)d_out;                // 4096 x 64

  // workspace (floats): h | s1 | s2 | rowmax | rowsum | partial[JSPLIT]
  float* h       = (float*)d_ws;
  float* s1      = h + (size_t)N_NODES * OUT_DIM;
  float* s2      = s1 + N_NODES;
  float* rowmax  = s2 + N_NODES;
  float* rowsum  = rowmax + N_NODES;
  float* partial = rowsum + N_NODES;         // JSPLIT * 4096 * 64 floats

  gat_h_gemm<<<N_NODES / 16, 128, 0, stream>>>(X, W, h);
  gat_scores<<<N_NODES / 256, 256, 0, stream>>>(h, a, s1, s2);
  gat_softmax_stats<<<N_NODES / 8, 256, 0, stream>>>(adj, s1, s2, rowmax,
                                                     rowsum);
  gat_attn_gemm<<<dim3(N_NODES / 64, JSPLIT), 128, 0, stream>>>(
      adj, s1, s2, rowmax, rowsum, h, partial);
  gat_reduce<<<(N_NODES * OUT_DIM + 255) / 256, 256, 0, stream>>>(partial,
                                                                  out);
}